// ConditionalMLN_71279277244794
// MI455X (gfx1250) — compile-verified
//
#include <hip/hip_runtime.h>

// Problem constants from the reference: R=16 rules, G=200000 groundings, K=3.
#define RR 16
#define GG 200000

typedef __attribute__((ext_vector_type(2))) float v2f;
typedef __attribute__((ext_vector_type(8))) float v8f;

// 12-byte index/mask tuple; aligned(4) so clang can merge into global_load_b96.
struct __attribute__((aligned(4))) I3 { int x, y, z; };

// ---------------------------------------------------------------------------
// Main kernel: memory-bound streaming reduction.
//   total[r,g] = (s0*s1*s2) - flag * (q0*q1*q2),  sk=(1-pk)+pk, qk = mk?pk:1-pk
//   contrib    = w[r] * total[r,g]
// Per-block partial sums go to d_ws (no float atomics -> deterministic).
// ---------------------------------------------------------------------------
__global__ __launch_bounds__(256) void mln_main(
    const float* __restrict__ post,          // posterior_prob (N)
    const float* __restrict__ w,             // rule_weights (R)
    const int* __restrict__ inds,            // latent_var_inds (R*G*3)
    const int* __restrict__ mask,            // latent_neg_mask (R*G*3)
    const unsigned char* __restrict__ flag,  // obs_zero_flag (R*G), 1 byte each
    float* __restrict__ partials)            // one float per block
{
    __shared__ float sw[RR];
    __shared__ float wave_sums[8];
    if (threadIdx.x < RR) sw[threadIdx.x] = w[threadIdx.x];
    __syncthreads();

    const int total = RR * GG;                       // 3,200,000 groups
    const int step  = gridDim.x * blockDim.x;
    int idx = blockIdx.x * blockDim.x + threadIdx.x;

    float acc = 0.0f;
    for (; idx < total; idx += step) {
        const long b = (long)idx * 3;

        // Prefetch next grid-stride chunk of the two hot streams
        // (lowers to gfx1250 global_prefetch_b8; speculative, safe to drop).
        if (idx + step < total) {
            __builtin_prefetch(inds + b + (long)step * 3, 0, 1);
            __builtin_prefetch(mask + b + (long)step * 3, 0, 1);
        }

        const I3 iv = *reinterpret_cast<const I3*>(inds + b);  // b96 load
        const I3 mv = *reinterpret_cast<const I3*>(mask + b);  // b96 load

        // 3 gathers; posterior_prob (8 MB) is L2-resident on MI455X (192 MB L2).
        const float p0 = post[iv.x], p1 = post[iv.y], p2 = post[iv.z];
        const float o0 = 1.0f - p0,  o1 = 1.0f - p1,  o2 = 1.0f - p2;

        const float q0 = mv.x ? p0 : o0;
        const float q1 = mv.y ? p1 : o1;
        const float q2 = mv.z ? p2 : o2;

        // Sum of the full 2^3 cartesian product == product of pair-sums.
        const float prodS = (o0 + p0) * (o1 + p1) * (o2 + p2);
        const float zeroed = q0 * q1 * q2;
        const float tot = prodS - (flag[idx] ? zeroed : 0.0f);

        const int r = idx / GG;                      // magic-number division
        acc += sw[r] * tot;
    }

    // wave32 shuffle reduction, then cross-wave via LDS.
    #pragma unroll
    for (int off = 16; off > 0; off >>= 1)
        acc += __shfl_down(acc, off, 32);

    const int wid  = threadIdx.x >> 5;
    const int lane = threadIdx.x & 31;
    if (lane == 0) wave_sums[wid] = acc;
    __syncthreads();

    if (wid == 0) {
        float v = (lane < 8) ? wave_sums[lane] : 0.0f;
        #pragma unroll
        for (int off = 4; off > 0; off >>= 1)
            v += __shfl_down(v, off, 32);
        if (lane == 0) partials[blockIdx.x] = v;
    }
}

// ---------------------------------------------------------------------------
// Epilogue: one wave reduces the per-block partials + the constant term
// sum_r w[r]*obs[r]. The final 32-lane -> scalar step goes through the FP32
// matrix pipe: one V_WMMA_F32_16X16X4_F32 with A = lane accumulators laid out
// as 16x4 (lanes 0-15 -> A[m,0], lanes 16-31 -> A[m,2]), B = ones, so
// D[m,n] = acc_m + acc_{m+16}; 8 per-lane adds + one shfl_xor(16) finish it.
// FP32 WMMA is required (f16 would overflow: per-rule scores ~2e5 > 65504).
// ---------------------------------------------------------------------------
__global__ __launch_bounds__(32) void mln_final(
    const float* __restrict__ partials, int nb,
    const float* __restrict__ w,
    const float* __restrict__ obs,
    float* __restrict__ out)
{
    const int lane = threadIdx.x;

    float acc = 0.0f;
    for (int i = lane; i < nb; i += 32) acc += partials[i];
    if (lane < RR) acc += w[lane] * obs[lane];
    // control flow reconverges here; EXEC is all-ones for the WMMA below.

    v2f a; a.x = acc;  a.y = 0.0f;   // A: 16x4, K=1 and K=3 columns zero
    v2f bm; bm.x = 1.0f; bm.y = 1.0f; // B: 4x16 all-ones
    v8f c = {};
    c = __builtin_amdgcn_wmma_f32_16x16x4_f32(
            /*neg_a=*/false, a, /*neg_b=*/false, bm,
            /*c_mod=*/(short)0, c, /*reuse_a=*/false, /*reuse_b=*/false);

    // D VGPR j: lanes 0-15 hold M=j, lanes 16-31 hold M=j+8 (value replicated
    // across N). Summing the 8 VGPRs gives M=0..7 on lanes 0-15 and M=8..15 on
    // lanes 16-31; shfl_xor(16) combines the halves.
    float s = c[0] + c[1] + c[2] + c[3] + c[4] + c[5] + c[6] + c[7];
    s += __shfl_xor(s, 16, 32);

    if (lane == 0) out[0] = s;
}

extern "C" void kernel_launch(void* const* d_in, const int* in_sizes, int n_in,
                              void* d_out, int out_size, void* d_ws, size_t ws_size,
                              hipStream_t stream) {
    (void)in_sizes; (void)n_in; (void)out_size; (void)ws_size;

    const float*         post = (const float*)d_in[0];         // posterior_prob
    const float*         obs  = (const float*)d_in[1];         // observed_rule_cnts
    const float*         w    = (const float*)d_in[2];         // rule_weights
    const int*           inds = (const int*)d_in[3];           // latent_var_inds
    const int*           mask = (const int*)d_in[4];           // latent_neg_mask
    const unsigned char* flag = (const unsigned char*)d_in[5]; // obs_zero_flag (bool)

    float* out      = (float*)d_out;
    float* partials = (float*)d_ws;   // NB floats of scratch

    const int NB = 2048;  // 2048 blocks x 256 thr = 512K threads, ~6 groups each
    mln_main<<<NB, 256, 0, stream>>>(post, w, inds, mask, flag, partials);
    mln_final<<<1, 32, 0, stream>>>(partials, NB, w, obs, out);
}